// MarcusGATConv_20255065768575
// MI455X (gfx1250) — compile-verified
//
#include <hip/hip_runtime.h>
#include <math.h>

#define FEATS 128   // IN_FEATS == OUT_FEATS == 128

typedef __attribute__((ext_vector_type(2))) float v2f;
typedef __attribute__((ext_vector_type(8))) float v8f;

__device__ __forceinline__ void atomic_add_f32(float* p, float v) {
    // lowers to global_atomic_add_f32 (no-return form since result unused)
    __hip_atomic_fetch_add(p, v, __ATOMIC_RELAXED, __HIP_MEMORY_SCOPE_AGENT);
}

// -----------------------------------------------------------------------------
// f = relu(feat[ids[.]] @ W^T + b)  with V_WMMA_F32_16X16X4_F32.
//
// W (64 KB) is staged once per block into LDS, quad-packed as
//   wlds[q*512 + n*4 + j] = W[n][4q + j]        (q = k/4, j = 0..3)
// so a B tile read is ONE ds_load_b64 per lane at  q*512 + n*4 + lhalf*2:
// lanes 0-15 hit bank pairs {0,1},{4,5},... and lanes 16-31 (offset +2 floats)
// hit {2,3},{6,7},... -> conflict-free, no padding, exactly 64 KB.
//
// One wave computes a 16(M) x 128(N) tile: 8 v8f accumulators, K-loop step 4.
// All 8 B tiles are loaded into distinct registers BEFORE the 8 WMMAs so the
// loads pipeline (one dscnt wait per K step instead of 8 load->wait->wmma).
//
// A layout (16x4 f32): lanes 0-15 hold K={k0,k0+1}, lanes 16-31 K={k0+2,k0+3}.
// C/D layout: VGPR r -> row r (lanes 0-15) and row r+8 (lanes 16-31).
// -----------------------------------------------------------------------------
__global__ void gemm_relu_wmma(const float* __restrict__ feat,
                               const int*   __restrict__ ids,
                               const float* __restrict__ W,    // [128,128] (OUT,IN)
                               const float* __restrict__ bias,
                               float*       __restrict__ outf, // [n_rows,128]
                               int n_rows)
{
    __shared__ float wlds[FEATS * FEATS];          // 64 KB, quad-packed

    // cooperative stage: idx -> (n = idx>>5, q = idx&31); global side is
    // perfectly coalesced 16B/thread.
    for (int idx = threadIdx.x; idx < FEATS * 32; idx += blockDim.x) {
        const int n = idx >> 5;
        const int q = idx & 31;
        const float4 w4 = *(const float4*)(W + (size_t)n * FEATS + q * 4);
        *(float4*)(wlds + q * 512 + n * 4) = w4;
    }
    __syncthreads();                               // all waves reach this

    const int lane  = threadIdx.x & 31;
    const int wave  = threadIdx.x >> 5;
    const int wpb   = blockDim.x >> 5;
    const int tile  = blockIdx.x * wpb + wave;     // 16-row tile index
    if (tile * 16 >= n_rows) return;               // wave-uniform exit

    const int row0  = tile * 16;
    const int l15   = lane & 15;
    const int lhalf = lane >> 4;                   // 0 or 1

    const int    srcrow = ids[row0 + l15];         // gather node -> feat row
    const float* aptr   = feat + (size_t)srcrow * FEATS + lhalf * 2;
    const float* bbase  = wlds + l15 * 4 + lhalf * 2;

    v8f acc[8] = {};                               // 8 N-tiles of 16

    for (int k0 = 0; k0 < FEATS; k0 += 4) {
        const v2f a = *(const v2f*)(aptr + k0);    // one global_load_b64
        const float* brow = bbase + (k0 >> 2) * 512;
        v2f bt[8];
#pragma unroll
        for (int nb = 0; nb < 8; ++nb)             // 8 ds_load_b64, pipelined
            bt[nb] = *(const v2f*)(brow + nb * 64);
#pragma unroll
        for (int nb = 0; nb < 8; ++nb)
#if __has_builtin(__builtin_amdgcn_wmma_f32_16x16x4_f32)
            acc[nb] = __builtin_amdgcn_wmma_f32_16x16x4_f32(
                false, a, false, bt[nb], (short)0, acc[nb], false, false);
#else
            acc[nb][0] += a[0] * bt[nb][0] + a[1] * bt[nb][1];
#endif
    }

    // bias + relu + store (D layout: VGPR r -> rows r / r+8)
#pragma unroll
    for (int nb = 0; nb < 8; ++nb) {
        const int n = nb * 16 + l15;
        const float bv = bias[n];
#pragma unroll
        for (int r = 0; r < 8; ++r) {
            const int m = row0 + r + lhalf * 8;
            float v = acc[nb][r] + bv;
            outf[(size_t)m * FEATS + n] = v > 0.f ? v : 0.f;
        }
    }
}

// -----------------------------------------------------------------------------
// Edge dot products: one wave per edge (grid-stride). Each lane loads a float4
// of both endpoint rows (512B/row, fully coalesced, L2-resident), xor-shuffle
// reduce over the 32 lanes (wave32). Tracks per-block max for global softmax.
// -----------------------------------------------------------------------------
__global__ void edge_dot_kernel(const float* __restrict__ feat,
                                const int*   __restrict__ user_ids,
                                const int*   __restrict__ item_ids,
                                const int*   __restrict__ esrc,
                                const int*   __restrict__ edst,
                                float*       __restrict__ alpha_raw,
                                float*       __restrict__ block_max,
                                int n_edges)
{
    __shared__ float smax[8];
    const int lane = threadIdx.x & 31;
    const int wave = threadIdx.x >> 5;
    const int wpb  = blockDim.x >> 5;
    const int gw   = blockIdx.x * wpb + wave;
    const int nw   = gridDim.x * wpb;
    const float scale = 0.08838834764831845f;      // 1/sqrt(128)

    float wmax = -INFINITY;
    for (int e = gw; e < n_edges; e += nw) {
        const int srow = user_ids[esrc[e]];
        const int drow = item_ids[edst[e]];
        const float4 a = ((const float4*)(feat + (size_t)srow * FEATS))[lane];
        const float4 b = ((const float4*)(feat + (size_t)drow * FEATS))[lane];
        float p = a.x * b.x + a.y * b.y + a.z * b.z + a.w * b.w;
#pragma unroll
        for (int off = 16; off > 0; off >>= 1) p += __shfl_xor(p, off, 32);
        p *= scale;
        if (lane == 0) alpha_raw[e] = p;
        wmax = fmaxf(wmax, p);
    }
    if (lane == 0) smax[wave] = wmax;
    __syncthreads();
    if (threadIdx.x == 0) {
        float m = smax[0];
        for (int i = 1; i < wpb; ++i) m = fmaxf(m, smax[i]);
        block_max[blockIdx.x] = m;
    }
}

// Single-block final reduction (is_sum: 1 = sum, 0 = max)
__global__ void reduce_final(const float* __restrict__ in, float* __restrict__ out,
                             int n, int is_sum)
{
    __shared__ float s[512];
    const int t = threadIdx.x;
    float v = is_sum ? 0.f : -INFINITY;
    for (int i = t; i < n; i += blockDim.x)
        v = is_sum ? (v + in[i]) : fmaxf(v, in[i]);
    s[t] = v;
    __syncthreads();
    for (int o = blockDim.x >> 1; o > 0; o >>= 1) {
        if (t < o) s[t] = is_sum ? (s[t] + s[t + o]) : fmaxf(s[t], s[t + o]);
        __syncthreads();
    }
    if (t == 0) out[0] = s[0];
}

// alpha[e] = exp(raw[e] - gmax), per-block partial sums
__global__ void exp_sum_kernel(float* __restrict__ alpha,
                               const float* __restrict__ scalars,   // [0] = gmax
                               float* __restrict__ block_sum,
                               int n_edges)
{
    __shared__ float ssum[256];
    const float m = scalars[0];
    float acc = 0.f;
    for (int i = blockIdx.x * blockDim.x + threadIdx.x; i < n_edges;
         i += gridDim.x * blockDim.x) {
        const float a = expf(alpha[i] - m);
        alpha[i] = a;
        acc += a;
    }
    ssum[threadIdx.x] = acc;
    __syncthreads();
    for (int o = blockDim.x >> 1; o > 0; o >>= 1) {
        if (threadIdx.x < o) ssum[threadIdx.x] += ssum[threadIdx.x + o];
        __syncthreads();
    }
    if (threadIdx.x == 0) block_sum[blockIdx.x] = ssum[0];
}

// -----------------------------------------------------------------------------
// Scatter: one wave per edge. w = alpha[e]/gsum; lane handles 4 features.
// e_item[d] += w * f_src[s]  (rows n_users..), e_user[s] += w * f_dst[d].
// All traffic is L2-resident (f buffers + out = ~154 MB < 192 MB L2).
// -----------------------------------------------------------------------------
__global__ void scatter_kernel(const float* __restrict__ alpha,
                               const float* __restrict__ scalars,   // [1] = gsum
                               const float* __restrict__ f_src,
                               const float* __restrict__ f_dst,
                               const int*   __restrict__ esrc,
                               const int*   __restrict__ edst,
                               float*       __restrict__ out,
                               int n_edges, int n_users)
{
    const int lane = threadIdx.x & 31;
    const int wave = threadIdx.x >> 5;
    const int e    = blockIdx.x * (blockDim.x >> 5) + wave;
    if (e >= n_edges) return;

    const float w = alpha[e] / scalars[1];
    const int s = esrc[e];
    const int d = edst[e];

    const float4 fs = ((const float4*)(f_src + (size_t)s * FEATS))[lane];
    const float4 fd = ((const float4*)(f_dst + (size_t)d * FEATS))[lane];

    float* oi = out + ((size_t)(n_users + d)) * FEATS + lane * 4;
    float* ou = out + (size_t)s * FEATS + lane * 4;

    atomic_add_f32(oi + 0, w * fs.x);
    atomic_add_f32(oi + 1, w * fs.y);
    atomic_add_f32(oi + 2, w * fs.z);
    atomic_add_f32(oi + 3, w * fs.w);
    atomic_add_f32(ou + 0, w * fd.x);
    atomic_add_f32(ou + 1, w * fd.y);
    atomic_add_f32(ou + 2, w * fd.z);
    atomic_add_f32(ou + 3, w * fd.w);
}

extern "C" void kernel_launch(void* const* d_in, const int* in_sizes, int n_in,
                              void* d_out, int out_size, void* d_ws, size_t ws_size,
                              hipStream_t stream) {
    const float* feat     = (const float*)d_in[0];
    const float* W_src    = (const float*)d_in[1];
    const float* b_src    = (const float*)d_in[2];
    const float* W_dst    = (const float*)d_in[3];
    const float* b_dst    = (const float*)d_in[4];
    const int*   user_ids = (const int*)d_in[5];
    const int*   item_ids = (const int*)d_in[6];
    const int*   edge_src = (const int*)d_in[7];
    const int*   edge_dst = (const int*)d_in[8];

    const int n_users = in_sizes[5];
    const int n_items = in_sizes[6];
    const int n_edges = in_sizes[7];

    // workspace layout (floats): f_src | f_dst | alpha | blockred | scalars
    const int RED = 512;
    float* f_src    = (float*)d_ws;
    float* f_dst    = f_src + (size_t)n_users * FEATS;
    float* alpha    = f_dst + (size_t)n_items * FEATS;
    float* blockred = alpha + (size_t)n_edges;
    float* scalars  = blockred + RED;              // [0]=gmax, [1]=gsum

    // zero output (atomic scatter accumulates into it)
    hipMemsetAsync(d_out, 0, (size_t)out_size * sizeof(float), stream);

    // projections: f = relu(h @ W^T + b), WMMA f32 16x16x4, W staged in LDS
    {
        dim3 blk(256);                             // 8 waves -> 128 rows/block
        const int tiles_u = (n_users + 15) / 16;
        gemm_relu_wmma<<<dim3((tiles_u + 7) / 8), blk, 0, stream>>>(
            feat, user_ids, W_src, b_src, f_src, n_users);
        const int tiles_i = (n_items + 15) / 16;
        gemm_relu_wmma<<<dim3((tiles_i + 7) / 8), blk, 0, stream>>>(
            feat, item_ids, W_dst, b_dst, f_dst, n_items);
    }

    // edge dots + global softmax (4 small passes)
    edge_dot_kernel<<<RED, 256, 0, stream>>>(feat, user_ids, item_ids,
                                             edge_src, edge_dst,
                                             alpha, blockred, n_edges);
    reduce_final<<<1, 512, 0, stream>>>(blockred, scalars + 0, RED, 0);
    exp_sum_kernel<<<RED, 256, 0, stream>>>(alpha, scalars, blockred, n_edges);
    reduce_final<<<1, 512, 0, stream>>>(blockred, scalars + 1, RED, 1);

    // weighted scatter-add into both endpoints
    scatter_kernel<<<dim3((n_edges + 7) / 8), 256, 0, stream>>>(
        alpha, scalars, f_src, f_dst, edge_src, edge_dst,
        (float*)d_out, n_edges, n_users);
}